// SelfGAT_83726092468499
// MI455X (gfx1250) — compile-verified
//
#include <hip/hip_runtime.h>
#include <hip/hip_bf16.h>

// ---------------------------------------------------------------------------
// GATv2Conv forward for MI455X (gfx1250, wave32).
//   x_l = feats @ W_l + b_l ; x_r = feats @ W_r + b_r      (WMMA f32 16x16x4,
//        A tile staged via global_load_async_to_lds_b128, B from W^T)
//   score_e = att . leakyrelu(x_l[src] + x_r[dst])          (wave per edge)
//   segment softmax over dst, out[dst] += alpha * x_l[src]  (L2 fp32 atomics)
// ---------------------------------------------------------------------------

typedef __attribute__((ext_vector_type(2))) float v2f;
typedef __attribute__((ext_vector_type(8))) float v8f;

#define DCH   256   // in channels
#define HOUT  256   // H*C out channels
#define HEADS 4
#define NEG_SLOPE 0.2f

__device__ __forceinline__ float lrelu(float x) { return x > 0.f ? x : NEG_SLOPE * x; }

// order-preserving float <-> uint map (for atomicMax-based segment max)
__device__ __forceinline__ unsigned fenc(float x) {
    unsigned u = __float_as_uint(x);
    return (u & 0x80000000u) ? ~u : (u | 0x80000000u);
}
__device__ __forceinline__ float fdec(unsigned u) {
    return (u & 0x80000000u) ? __uint_as_float(u & 0x7FFFFFFFu) : __uint_as_float(~u);
}
#define ENC_NEG_INF 0x007FFFFFu   // fenc(-inf)

// ---------------------------------------------------------------------------
// 256x256 transpose (32x32 LDS tiles, 256 threads = 32x8)
// ---------------------------------------------------------------------------
__global__ __launch_bounds__(256) void gatv2_transpose256(
    const float* __restrict__ W, float* __restrict__ WT)
{
    __shared__ float tile[32][33];
    const int bx = blockIdx.x * 32, by = blockIdx.y * 32;
    const int tx = threadIdx.x & 31, ty = threadIdx.x >> 5;
    #pragma unroll
    for (int j = 0; j < 32; j += 8)
        tile[ty + j][tx] = W[(size_t)(by + ty + j) * 256 + bx + tx];
    __syncthreads();
    #pragma unroll
    for (int j = 0; j < 32; j += 8)
        WT[(size_t)(bx + ty + j) * 256 + by + tx] = tile[tx][ty + j];
}

// ---------------------------------------------------------------------------
// Kernel 1: fused dual GEMM (blockIdx.y selects W_l/W_r), B taken from W^T.
// Block = 256 threads = 8 waves; block computes 16 rows x 256 cols.
// Wave w computes two 16x16 tiles at columns [32w, 32w+32).
// A-frag (f32 16x4): lane m (0-15): K = {0,1}; lane m+16: K = {2,3}.
// B-frag mirrored:   VGPR j holds B[K = 2*(lane/16)+j][N = lane%16].
// A tile staged memory->LDS with async DMA (ASYNCcnt).
// ---------------------------------------------------------------------------
__global__ __launch_bounds__(256) void gatv2_gemm_wmma(
    const float* __restrict__ feats,
    const float* __restrict__ WTl, const float* __restrict__ bl,
    const float* __restrict__ WTr, const float* __restrict__ br,
    float* __restrict__ xl, float* __restrict__ xr, int N)
{
    const float* __restrict__ WT   = (blockIdx.y == 0) ? WTl : WTr;
    const float* __restrict__ bias = (blockIdx.y == 0) ? bl : br;
    float* __restrict__ out        = (blockIdx.y == 0) ? xl : xr;

    __shared__ float As[16 * 260];           // 16 rows x 256 cols, pad to 260
    const int tid = threadIdx.x;
    const int rowBase = blockIdx.x * 16;

    // ---- async DMA stage of the 16x256 A tile (4 x 16B per thread) --------
    // row stride 260 floats = 1040 B = 65*16 -> every float4 slot 16B aligned
    const unsigned ldsBase = (unsigned)(uintptr_t)(void*)As;  // LDS offset
    for (int i = tid; i < 1024; i += 256) {
        int r  = i >> 6;           // row within tile
        int c4 = i & 63;           // float4 column
        int rr = rowBase + r; if (rr >= N) rr = N - 1;
        const float* gsrc = feats + (size_t)rr * DCH + c4 * 4;
        unsigned ldst = ldsBase + (unsigned)(r * 260 + c4 * 4) * 4u;
        asm volatile("global_load_async_to_lds_b128 %0, %1, off"
                     :: "v"(ldst), "v"((unsigned long long)(uintptr_t)gsrc)
                     : "memory");
    }
    asm volatile("s_wait_asynccnt 0x0" ::: "memory");
    __syncthreads();

    const int lane = tid & 31;
    const int wave = tid >> 5;
    const int m  = lane & 15;      // matrix row (A) / column (B)
    const int kh = lane >> 4;      // K half-select: 0 -> K={0,1}, 1 -> K={2,3}
    const int col0 = wave * 32;

    // W^T rows for this lane's two output columns (contiguous in K)
    const v2f* __restrict__ wt0 = (const v2f*)(WT + (size_t)(col0 + m) * DCH);
    const v2f* __restrict__ wt1 = (const v2f*)(WT + (size_t)(col0 + 16 + m) * DCH);

    v8f acc0 = {}; v8f acc1 = {};
    for (int kk = 0; kk < DCH; kk += 4) {
        const int k = kk + 2 * kh;
        v2f a  = *(const v2f*)(&As[m * 260 + k]);   // 8B-aligned LDS load
        v2f b0 = wt0[k >> 1];                       // single b64 per tile
        v2f b1 = wt1[k >> 1];
        acc0 = __builtin_amdgcn_wmma_f32_16x16x4_f32(false, a, false, b0,
                                                     (short)0, acc0, false, false);
        acc1 = __builtin_amdgcn_wmma_f32_16x16x4_f32(false, a, false, b1,
                                                     (short)0, acc1, false, false);
    }

    // C/D layout: VGPR v, lanes 0-15 -> row v, lanes 16-31 -> row v+8
    #pragma unroll
    for (int v = 0; v < 8; ++v) {
        int row = rowBase + v + 8 * kh;
        if (row < N) {
            int c0 = col0 + m, c1 = col0 + 16 + m;
            out[(size_t)row * HOUT + c0] = acc0[v] + bias[c0];
            out[(size_t)row * HOUT + c1] = acc1[v] + bias[c1];
        }
    }
}

// ---------------------------------------------------------------------------
// Init: out = bias (broadcast), nodeMax = enc(-inf), denom = 0
// ---------------------------------------------------------------------------
__global__ void gatv2_init_out(float* __restrict__ out, const float* __restrict__ bias, int N) {
    int t = blockIdx.x * blockDim.x + threadIdx.x;
    if (t < N * HOUT) out[t] = bias[t & (HOUT - 1)];
}
__global__ void gatv2_init_node(unsigned* __restrict__ nodeMax, float* __restrict__ denom, int N) {
    int t = blockIdx.x * blockDim.x + threadIdx.x;
    if (t < N * HEADS) { nodeMax[t] = ENC_NEG_INF; denom[t] = 0.f; }
}

// ---------------------------------------------------------------------------
// Kernel 2: per-edge scores + segment max. One wave32 per edge.
// Lane l owns channels [8l, 8l+8); head = l>>3; reduce within 8-lane group.
// ---------------------------------------------------------------------------
__global__ __launch_bounds__(256) void gatv2_scores(
    const float* __restrict__ xl, const float* __restrict__ xr,
    const int* __restrict__ edges, const float* __restrict__ att,
    float* __restrict__ scores, unsigned* __restrict__ nodeMax, int E, int ET)
{
    const int wave = (blockIdx.x * blockDim.x + threadIdx.x) >> 5;
    const int lane = threadIdx.x & 31;
    if (wave >= ET) return;
    int src, dst;
    if (wave < E) { src = edges[wave]; dst = edges[E + wave]; }
    else          { src = dst = wave - E; }

    const float4* xls = (const float4*)(xl + (size_t)src * HOUT);
    const float4* xrd = (const float4*)(xr + (size_t)dst * HOUT);
    const float4* at  = (const float4*)att;   // att is (H,C) flat = 256 floats

    float4 a0 = xls[lane * 2], a1 = xls[lane * 2 + 1];
    float4 b0 = xrd[lane * 2], b1 = xrd[lane * 2 + 1];
    float4 w0 = at [lane * 2], w1 = at [lane * 2 + 1];

    float p = 0.f;
    p += w0.x * lrelu(a0.x + b0.x);
    p += w0.y * lrelu(a0.y + b0.y);
    p += w0.z * lrelu(a0.z + b0.z);
    p += w0.w * lrelu(a0.w + b0.w);
    p += w1.x * lrelu(a1.x + b1.x);
    p += w1.y * lrelu(a1.y + b1.y);
    p += w1.z * lrelu(a1.z + b1.z);
    p += w1.w * lrelu(a1.w + b1.w);

    // sum across the 8 lanes of this head group
    p += __shfl_xor(p, 1);
    p += __shfl_xor(p, 2);
    p += __shfl_xor(p, 4);

    if ((lane & 7) == 0) {
        int h = lane >> 3;
        scores[(size_t)wave * HEADS + h] = p;
        atomicMax(&nodeMax[dst * HEADS + h], fenc(p));
    }
}

// ---------------------------------------------------------------------------
// Kernel 3: ex = exp(score - max[dst]); denom[dst] += ex. Thread per (edge, head).
// ---------------------------------------------------------------------------
__global__ __launch_bounds__(256) void gatv2_exp(
    const int* __restrict__ edges, float* __restrict__ scores,
    const unsigned* __restrict__ nodeMax, float* __restrict__ denom, int E, int ET)
{
    int t = blockIdx.x * blockDim.x + threadIdx.x;
    if (t >= ET * HEADS) return;
    int e = t >> 2, h = t & 3;
    int dst = (e < E) ? edges[E + e] : e - E;
    float m  = fdec(nodeMax[dst * HEADS + h]);
    float ex = __expf(scores[t] - m);
    scores[t] = ex;
    unsafeAtomicAdd(&denom[dst * HEADS + h], ex);
}

// ---------------------------------------------------------------------------
// Kernel 4: out[dst] += (ex/denom[dst]) * x_l[src]. One wave32 per edge.
// ---------------------------------------------------------------------------
__global__ __launch_bounds__(256) void gatv2_aggregate(
    const float* __restrict__ xl, const int* __restrict__ edges,
    const float* __restrict__ scores, const float* __restrict__ denom,
    float* __restrict__ out, int E, int ET)
{
    const int wave = (blockIdx.x * blockDim.x + threadIdx.x) >> 5;
    const int lane = threadIdx.x & 31;
    if (wave >= ET) return;
    int src, dst;
    if (wave < E) { src = edges[wave]; dst = edges[E + wave]; }
    else          { src = dst = wave - E; }

    const float* xrow = xl + (size_t)src * HOUT;
    __builtin_prefetch(xrow + lane * 8, 0, 0);          // global_prefetch_b8

    const int h = lane >> 3;
    float ex  = scores[(size_t)wave * HEADS + h];
    float den = denom[dst * HEADS + h];
    float alpha = ex / (den + 1e-16f);

    const float4* xls = (const float4*)xrow;
    float4 a0 = xls[lane * 2], a1 = xls[lane * 2 + 1];
    float* o = out + (size_t)dst * HOUT + lane * 8;
    unsafeAtomicAdd(o + 0, alpha * a0.x);
    unsafeAtomicAdd(o + 1, alpha * a0.y);
    unsafeAtomicAdd(o + 2, alpha * a0.z);
    unsafeAtomicAdd(o + 3, alpha * a0.w);
    unsafeAtomicAdd(o + 4, alpha * a1.x);
    unsafeAtomicAdd(o + 5, alpha * a1.y);
    unsafeAtomicAdd(o + 6, alpha * a1.z);
    unsafeAtomicAdd(o + 7, alpha * a1.w);
}

// ---------------------------------------------------------------------------
extern "C" void kernel_launch(void* const* d_in, const int* in_sizes, int n_in,
                              void* d_out, int out_size, void* d_ws, size_t ws_size,
                              hipStream_t stream) {
    const float* feats = (const float*)d_in[0];
    const int*   edges = (const int*)  d_in[1];
    // d_in[2] = batches (unused: single graph)
    const float* W_l  = (const float*)d_in[3];
    const float* b_l  = (const float*)d_in[4];
    const float* W_r  = (const float*)d_in[5];
    const float* b_r  = (const float*)d_in[6];
    const float* att  = (const float*)d_in[7];
    const float* bias = (const float*)d_in[8];
    float* out = (float*)d_out;

    const int N  = in_sizes[0] / DCH;
    const int E  = in_sizes[1] / 2;
    const int ET = E + N;            // with self loops

    // workspace carve-up (all offsets multiples of 16 bytes)
    float*    xl      = (float*)d_ws;
    float*    xr      = xl + (size_t)N * HOUT;
    float*    scores  = xr + (size_t)N * HOUT;
    unsigned* nodeMax = (unsigned*)(scores + (size_t)ET * HEADS);
    float*    denom   = (float*)(nodeMax + (size_t)N * HEADS);
    float*    WTl     = denom + (size_t)N * HEADS;
    float*    WTr     = WTl + (size_t)DCH * HOUT;

    // init
    gatv2_init_node<<<(N * HEADS + 255) / 256, 256, 0, stream>>>(nodeMax, denom, N);
    gatv2_init_out <<<(N * HOUT  + 255) / 256, 256, 0, stream>>>(out, bias, N);

    // weight transposes (so GEMM B-fragments are contiguous b64 loads)
    dim3 tgrid(8, 8);
    gatv2_transpose256<<<tgrid, 256, 0, stream>>>(W_l, WTl);
    gatv2_transpose256<<<tgrid, 256, 0, stream>>>(W_r, WTr);

    // dual GEMM: x_l, x_r (WMMA f32, async-LDS A staging)
    dim3 ggrid((N + 15) / 16, 2);
    gatv2_gemm_wmma<<<ggrid, 256, 0, stream>>>(feats, WTl, b_l, WTr, b_r, xl, xr, N);

    // edge phase
    int eblocks = (ET * 32 + 255) / 256;
    gatv2_scores<<<eblocks, 256, 0, stream>>>(xl, xr, edges, att, scores, nodeMax, E, ET);
    gatv2_exp<<<(ET * HEADS + 255) / 256, 256, 0, stream>>>(edges, scores, nodeMax, denom, E, ET);
    gatv2_aggregate<<<eblocks, 256, 0, stream>>>(xl, edges, scores, denom, out, E, ET);
}